// QuantizerTl_81896436400536
// MI455X (gfx1250) — compile-verified
//
#include <hip/hip_runtime.h>
#include <stdint.h>

// ---------------------------------------------------------------------------
// Per-row int4 dynamic quantizer for MI455X (gfx1250).
// x: (4,4096, 128+4096) f32.  Per row of 4224 floats:
//   outlier = x[0:128]                 (copied through verbatim)
//   absmax  = max|x[128:4224]|
//   scale   = max(absmax,1e-8) * 1.0/7.0
//   q       = clip(rint(v/scale), -8, 7)   (rint == round-half-even == jnp.round)
//   packed  = (q[0::2]&15) | (q[1::2]<<4)  (2048 bytes)
//
// Memory-bound: ~319 MB total traffic -> ~13.7us floor @ 23.3 TB/s.
// One 8-wave block per row; thread t owns elements [16t,16t+16) so its 8
// packed bytes are contiguous -> single b64 store per lane.
// Outlier copy uses the gfx1250 async global<->LDS engine (ASYNCcnt).
// ---------------------------------------------------------------------------

typedef float    v4f __attribute__((ext_vector_type(4)));
typedef uint32_t v2u __attribute__((ext_vector_type(2)));

#define POD            128
#define HID            4096
#define ROW_IN         (POD + HID)   /* 4224 floats per input row   */
#define PACKED_PER_ROW (HID / 2)     /* 2048 bytes per output row   */
#define QINV           (1.0f / 7.0f) /* CLIP/QMAX                   */
#define EPSQ           1e-8f

__global__ __launch_bounds__(256) void quantizer_tl_kernel(
    const float* __restrict__ x,
    int8_t*  __restrict__ packed_out,
    float*   __restrict__ scale_out,
    float*   __restrict__ outlier_out,
    int rows)
{
    const int row = blockIdx.x;
    if (row >= rows) return;
    const int t    = threadIdx.x;
    const int lane = t & 31;
    const int wave = t >> 5;

    const float* xrow = x + (size_t)row * ROW_IN;
    const float* xq   = xrow + POD;

    __shared__ float s_red[8];
    __shared__ float s_scale;
    __shared__ float s_inv;
    __shared__ __align__(16) float s_outlier[POD];

    // ---- kick off async DMA of the 512B outlier block: global -> LDS ------
    // wave 0 only: 32 lanes x b128 = 128 floats.  Tracked by ASYNCcnt.
    if (t < 32) {
        uint32_t lds_off = (uint32_t)(uintptr_t)(&s_outlier[lane * 4]);
        uint32_t goff    = (uint32_t)(lane * 16);  // byte offset from xrow
        asm volatile("global_load_async_to_lds_b128 %0, %1, %2"
                     :: "v"(lds_off), "v"(goff), "s"(xrow)
                     : "memory");
    }

    // ---- main data: 4x b128 non-temporal loads (streamed, bypass-friendly)
    const v4f* p = (const v4f*)(xq + t * 16);
    v4f v0 = __builtin_nontemporal_load(p + 0);
    v4f v1 = __builtin_nontemporal_load(p + 1);
    v4f v2 = __builtin_nontemporal_load(p + 2);
    v4f v3 = __builtin_nontemporal_load(p + 3);

    float vals[16];
    #pragma unroll
    for (int i = 0; i < 4; ++i) {
        vals[0  + i] = v0[i];
        vals[4  + i] = v1[i];
        vals[8  + i] = v2[i];
        vals[12 + i] = v3[i];
    }

    // ---- per-thread absmax over 16 values ---------------------------------
    float m = 0.0f;
    #pragma unroll
    for (int i = 0; i < 16; ++i)
        m = fmaxf(m, fabsf(vals[i]));

    // ---- wave32 butterfly max reduction -----------------------------------
    #pragma unroll
    for (int off = 16; off > 0; off >>= 1)
        m = fmaxf(m, __shfl_xor(m, off, 32));

    if (lane == 0) s_red[wave] = m;
    __syncthreads();

    if (t == 0) {
        float mm = s_red[0];
        #pragma unroll
        for (int i = 1; i < 8; ++i) mm = fmaxf(mm, s_red[i]);
        float sc = fmaxf(mm, EPSQ) * QINV;
        s_scale  = sc;
        s_inv    = 1.0f / sc;
        scale_out[row] = sc;
    }
    __syncthreads();

    const float inv = s_inv;

    // ---- quantize + nibble-pack: 16 values -> 8 bytes -> one b64 NT store -
    uint32_t w0 = 0, w1 = 0;
    #pragma unroll
    for (int i = 0; i < 8; ++i) {
        float rlo = rintf(vals[2 * i + 0] * inv);
        float rhi = rintf(vals[2 * i + 1] * inv);
        rlo = fminf(fmaxf(rlo, -8.0f), 7.0f);
        rhi = fminf(fmaxf(rhi, -8.0f), 7.0f);
        int qlo = (int)rlo;
        int qhi = (int)rhi;
        uint32_t byte = (uint32_t)((qlo & 0xF) | ((qhi & 0xF) << 4));
        if (i < 4) w0 |= byte << (8 * i);
        else       w1 |= byte << (8 * (i - 4));
    }
    v2u w = { w0, w1 };
    v2u* outp = (v2u*)(packed_out + (size_t)row * PACKED_PER_ROW + t * 8);
    __builtin_nontemporal_store(w, outp);

    // ---- drain outlier DMA: LDS -> global (async store, ASYNCcnt) ---------
    if (t < 32) {
        // wait for the async load so LDS holds valid data before the store
        asm volatile("s_wait_asynccnt 0x0" ::: "memory");
        uint32_t lds_off = (uint32_t)(uintptr_t)(&s_outlier[lane * 4]);
        uint32_t goff    = (uint32_t)(lane * 16);  // byte offset from orow
        const float* orow = outlier_out + (size_t)row * POD;
        asm volatile("global_store_async_from_lds_b128 %0, %1, %2"
                     :: "v"(goff), "v"(lds_off), "s"(orow)
                     : "memory");
        asm volatile("s_wait_asynccnt 0x0" ::: "memory");
    }
}

extern "C" void kernel_launch(void* const* d_in, const int* in_sizes, int n_in,
                              void* d_out, int out_size, void* d_ws, size_t ws_size,
                              hipStream_t stream)
{
    (void)n_in; (void)out_size; (void)d_ws; (void)ws_size;
    const float* x = (const float*)d_in[0];
    const int rows = in_sizes[0] / ROW_IN;  // 4*4096 = 16384

    // Outputs concatenated flat in return order:
    //   [0]                        packed : rows*2048 int8
    //   [rows*2048]                scale  : rows f32
    //   [rows*2048 + rows*4]       outlier: rows*128 f32
    char*  base        = (char*)d_out;
    size_t packedBytes = (size_t)rows * PACKED_PER_ROW;
    int8_t* packed  = (int8_t*)base;
    float*  scale   = (float*)(base + packedBytes);
    float*  outlier = (float*)(base + packedBytes + (size_t)rows * sizeof(float));

    quantizer_tl_kernel<<<rows, 256, 0, stream>>>(x, packed, scale, outlier, rows);
}